// NTXentLossChildren2_85624468013606
// MI455X (gfx1250) — compile-verified
//
#include <hip/hip_runtime.h>
#include <hip/hip_bf16.h>
#include <math.h>

// Problem constants from the reference
#define BB   64
#define JJ   8
#define KK   64
#define MM   32
#define DD   256
#define NNEG (KK * MM)        // 2048 negatives per batch
#define INV_TEMP 10.0f        // 1/0.1
#define EPSF 1e-8f

typedef float v2f __attribute__((ext_vector_type(2)));
typedef float v8f __attribute__((ext_vector_type(8)));

// ---------------------------------------------------------------------------
// Kernel A: per-batch pos_sim (already /TEMP) and 1/max(||child||, EPS).
// ws[b] = pos_sim[b], ws[64+b] = inv_norm_child[b]
// ---------------------------------------------------------------------------
__global__ __launch_bounds__(256)
void ntx_pos_kernel(const float* __restrict__ childrens,
                    const float* __restrict__ childrens_pos,
                    float* __restrict__ ws) {
    const int b = blockIdx.x;
    const int d = threadIdx.x;                  // 256 threads == D
    const float c = childrens[(b * JJ + (JJ - 1)) * DD + d];
    const float p = childrens_pos[(b * JJ + (JJ - 1)) * DD + d];
    float cc = c * c, pp = p * p, cp = c * p;
    #pragma unroll
    for (int off = 16; off; off >>= 1) {
        cc += __shfl_xor(cc, off);
        pp += __shfl_xor(pp, off);
        cp += __shfl_xor(cp, off);
    }
    __shared__ float sred[3][8];
    const int lane = threadIdx.x & 31, w = threadIdx.x >> 5;
    if (lane == 0) { sred[0][w] = cc; sred[1][w] = pp; sred[2][w] = cp; }
    __syncthreads();
    if (threadIdx.x == 0) {
        float a = 0.f, bb2 = 0.f, c2 = 0.f;
        #pragma unroll
        for (int i = 0; i < 8; ++i) { a += sred[0][i]; bb2 += sred[1][i]; c2 += sred[2][i]; }
        const float nc = fmaxf(sqrtf(a), EPSF);
        const float np = fmaxf(sqrtf(bb2), EPSF);
        ws[b]      = c2 / (nc * np) * INV_TEMP;
        ws[64 + b] = 1.0f / nc;
    }
}

// ---------------------------------------------------------------------------
// Kernel B: streaming pass over negatives. grid = (8 chunks, 64 batches),
// block = 256 threads (8 waves). Each wave computes 32 neg dot products via
// V_WMMA_F32_16X16X4_F32 (two groups of 16 rows), neg norms via rider FMAs,
// masks by label, and folds into an online (max, sumexp) partial.
// Partials at ws+128: float2 per (b, chunk).
// ---------------------------------------------------------------------------
__global__ __launch_bounds__(256)
void ntx_neg_kernel(const float* __restrict__ negs,
                    const int*   __restrict__ neg_labels,
                    const int*   __restrict__ gt_labels,
                    const float* __restrict__ childrens,
                    const float* __restrict__ ws_in,
                    float2* __restrict__ partials) {
    const int chunk = blockIdx.x;               // 0..7  (256 negs each)
    const int b     = blockIdx.y;               // 0..63
    const int tid   = threadIdx.x;
    const int lane  = tid & 31;
    const int wav   = tid >> 5;

    // Stage child vector in LDS (B-fragment source: all 16 columns identical).
    __shared__ float sChild[DD];
    sChild[tid] = childrens[(b * JJ + (JJ - 1)) * DD + tid];
    __syncthreads();

    const float inv_nc  = ws_in[64 + b];
    const int   myLabel = gt_labels[b * JJ + (JJ - 1)];

    const int colOff   = (lane >> 4) * 2;       // k-offset for this half-wave
    const int rowInGrp = lane & 15;             // A-matrix row M = lane%16
    const int n0       = chunk * 256 + wav * 32;

    const float* __restrict__ negs_b = negs + (size_t)b * NNEG * DD;
    const float* __restrict__ ap0 = negs_b + (size_t)(n0 + rowInGrp) * DD + colOff;
    const float* __restrict__ ap1 = ap0 + 16 * DD;

    v8f acc0 = {}; v8f acc1 = {};
    float sq0 = 0.f, sq1 = 0.f;

    #pragma unroll 4
    for (int t = 0; t < 64; ++t) {              // K = 256 = 64 * 4
        v2f a0 = *(const v2f*)(ap0 + 4 * t);
        v2f a1 = *(const v2f*)(ap1 + 4 * t);
        v2f bv = *(const v2f*)(&sChild[4 * t + colOff]);
        __builtin_prefetch(ap0 + 4 * t + 64, 0, 0);
        __builtin_prefetch(ap1 + 4 * t + 64, 0, 0);
        acc0 = __builtin_amdgcn_wmma_f32_16x16x4_f32(false, a0, false, bv,
                                                     (short)0, acc0, false, false);
        acc1 = __builtin_amdgcn_wmma_f32_16x16x4_f32(false, a1, false, bv,
                                                     (short)0, acc1, false, false);
        sq0 = fmaf(a0.x, a0.x, fmaf(a0.y, a0.y, sq0));
        sq1 = fmaf(a1.x, a1.x, fmaf(a1.y, a1.y, sq1));
    }

    // Online logsumexp over this wave's 32 negatives.
    float m = -INFINITY, s = 0.0f;
    const int* __restrict__ nl = neg_labels + b * NNEG;
    #pragma unroll
    for (int g = 0; g < 2; ++g) {
        float sq  = g ? sq1 : sq0;
        v8f   acc = g ? acc1 : acc0;
        sq = sq + __shfl_xor(sq, 16);           // full row sum-of-squares
        const int lab = nl[n0 + g * 16 + rowInGrp];
        #pragma unroll
        for (int r = 0; r < 8; ++r) {
            const int   mrow = r + ((lane >> 4) << 3);   // C layout: M = r + 8*(lane/16)
            const float sqr  = __shfl(sq, mrow);
            const int   labr = __shfl(lab, mrow);
            const float dot  = acc[r];
            const float nb   = fmaxf(sqrtf(sqr), EPSF);
            const float sim  = dot * inv_nc * INV_TEMP / nb;
            if (labr == myLabel) {
                if (sim > m) { s = s * __expf(m - sim) + 1.0f; m = sim; }
                else         { s += __expf(sim - m); }
            }
        }
    }

    // Reduce (m, s) across the wave.
    #pragma unroll
    for (int off = 16; off; off >>= 1) {
        const float om = __shfl_xor(m, off);
        const float os = __shfl_xor(s, off);
        if (om > m)              { s = s * __expf(m - om) + os; m = om; }
        else if (om > -INFINITY) { s += os * __expf(om - m); }
    }

    // Reduce across the 8 waves of the block.
    __shared__ float redM[8], redS[8];
    if (lane == 0) { redM[wav] = m; redS[wav] = s; }
    __syncthreads();
    if (tid == 0) {
        float M = redM[0], S = redS[0];
        #pragma unroll
        for (int i = 1; i < 8; ++i) {
            const float om = redM[i], os = redS[i];
            if (om > M)              { S = S * __expf(M - om) + os; M = om; }
            else if (om > -INFINITY) { S += os * __expf(om - M); }
        }
        partials[b * 8 + chunk] = make_float2(M, S);
    }
}

// ---------------------------------------------------------------------------
// Kernel C: combine partials + pos_sim, per-batch loss, global sum / (2B).
// ---------------------------------------------------------------------------
__global__ __launch_bounds__(64)
void ntx_finalize_kernel(const float* __restrict__ ws, float* __restrict__ out) {
    __shared__ float sl[64];
    const int b = threadIdx.x;                  // 64 threads
    const float pos = ws[b];
    float m = pos, s = 1.0f;                    // logits start with pos_sim
    const float2* __restrict__ p = (const float2*)(ws + 128);
    #pragma unroll
    for (int c = 0; c < 8; ++c) {
        const float2 v = p[b * 8 + c];
        const float om = v.x, os = v.y;
        if (om > m)              { s = s * __expf(m - om) + os; m = om; }
        else if (om > -INFINITY) { s += os * __expf(om - m); }
    }
    sl[b] = m + __logf(s) - pos;
    __syncthreads();
    if (b == 0) {
        float t = 0.f;
        #pragma unroll
        for (int i = 0; i < 64; ++i) t += sl[i];
        out[0] = t / (2.0f * BB);
    }
}

// ---------------------------------------------------------------------------
extern "C" void kernel_launch(void* const* d_in, const int* in_sizes, int n_in,
                              void* d_out, int out_size, void* d_ws, size_t ws_size,
                              hipStream_t stream) {
    const float* childrens       = (const float*)d_in[0];
    const float* childrens_pos   = (const float*)d_in[1];
    const float* childrens_negs  = (const float*)d_in[2];
    const int*   gt_labels       = (const int*)d_in[3];
    const int*   gt_label_negs   = (const int*)d_in[4];
    float* ws  = (float*)d_ws;   // [0..63]=pos_sim, [64..127]=inv_nc, [128..]=partials (64*8 float2)
    float* out = (float*)d_out;

    ntx_pos_kernel<<<BB, 256, 0, stream>>>(childrens, childrens_pos, ws);

    dim3 grid(8, BB);
    ntx_neg_kernel<<<grid, 256, 0, stream>>>(childrens_negs, gt_label_negs, gt_labels,
                                             childrens, ws, (float2*)(ws + 128));

    ntx_finalize_kernel<<<1, 64, 0, stream>>>(ws, out);
}